// GaussianModel3D_23862838296740
// MI455X (gfx1250) — compile-verified
//
#include <hip/hip_runtime.h>

// MI455X (gfx1250) — bandwidth-bound streaming kernel.
// cov = (R*diag(exp(s))) @ (R*diag(exp(s)))^T per Gaussian, 4M items.
// Round 3: fix builtin signature — async global<->LDS builtins take
// (v4i addrspace(1)*, v4i addrspace(3)*, imm offset, imm cpol).

#define TPB 256

typedef float v4f __attribute__((ext_vector_type(4)));
typedef int   v4i __attribute__((ext_vector_type(4)));

#define GAS __attribute__((address_space(1)))
#define LAS __attribute__((address_space(3)))

#if defined(__HIP_DEVICE_COMPILE__) && \
    __has_builtin(__builtin_amdgcn_global_load_async_to_lds_b128)
#define HAVE_ASYNC_LOAD 1
#else
#define HAVE_ASYNC_LOAD 0
#endif

#if defined(__HIP_DEVICE_COMPILE__) && \
    __has_builtin(__builtin_amdgcn_global_store_async_from_lds_b128)
#define HAVE_ASYNC_STORE 1
#else
#define HAVE_ASYNC_STORE 0
#endif

// ---- async global -> LDS, 16B per active lane --------------------------------
__device__ __forceinline__ void async_g2l_b128(const v4f* g, v4f* l) {
#if HAVE_ASYNC_LOAD
  v4i* gg = (v4i*)g;            // generic reinterpret (drops const for builtin)
  v4i* ll = (v4i*)l;
  __builtin_amdgcn_global_load_async_to_lds_b128(
      (GAS v4i*)gg, (LAS v4i*)ll, /*offset=*/0, /*cpol=*/1 /*TH_NT*/);
#else
  unsigned lds_off = (unsigned)(__UINTPTR_TYPE__)(LAS void*)(void*)l;
  unsigned long long ga = (unsigned long long)(__UINTPTR_TYPE__)g;
  asm volatile("global_load_async_to_lds_b128 %0, %1, off th:TH_LOAD_NT"
               :: "v"(lds_off), "v"(ga) : "memory");
#endif
}

// ---- async LDS -> global, 16B per active lane --------------------------------
__device__ __forceinline__ void async_l2g_b128(v4f* g, const v4f* l) {
#if HAVE_ASYNC_STORE
  v4i* gg = (v4i*)g;
  v4i* ll = (v4i*)l;            // generic reinterpret (drops const for builtin)
  __builtin_amdgcn_global_store_async_from_lds_b128(
      (GAS v4i*)gg, (LAS v4i*)ll, /*offset=*/0, /*cpol=*/1 /*TH_NT*/);
#else
  unsigned lds_off = (unsigned)(__UINTPTR_TYPE__)(LAS void*)(void*)l;
  unsigned long long ga = (unsigned long long)(__UINTPTR_TYPE__)g;
  asm volatile("global_store_async_from_lds_b128 %1, %0, off th:TH_STORE_NT"
               :: "v"(lds_off), "v"(ga) : "memory");
#endif
}

__device__ __forceinline__ void wait_async0() {
#if defined(__HIP_DEVICE_COMPILE__) && \
    __has_builtin(__builtin_amdgcn_s_wait_asynccnt)
  __builtin_amdgcn_s_wait_asynccnt(0);
#else
  asm volatile("s_wait_asynccnt 0x0" ::: "memory");
#endif
}

__device__ __forceinline__ void cov_from(v4f q, float s0, float s1, float s2,
                                         float c[9]) {
  float w = q.x, x = q.y, y = q.z, z = q.w;
  float nrm = sqrtf(w * w + x * x + y * y + z * z) + 1e-8f;
  float inv = 1.0f / nrm;
  w *= inv; x *= inv; y *= inv; z *= inv;

  float r00 = 1.0f - 2.0f * (y * y + z * z);
  float r01 = 2.0f * (x * y - z * w);
  float r02 = 2.0f * (x * z + y * w);
  float r10 = 2.0f * (x * y + z * w);
  float r11 = 1.0f - 2.0f * (x * x + z * z);
  float r12 = 2.0f * (y * z - x * w);
  float r20 = 2.0f * (x * z - y * w);
  float r21 = 2.0f * (y * z + x * w);
  float r22 = 1.0f - 2.0f * (x * x + y * y);

  float e0 = __expf(s0), e1 = __expf(s1), e2 = __expf(s2);
  float t0 = e0 * e0, t1 = e1 * e1, t2 = e2 * e2;

  c[0] = r00 * r00 * t0 + r01 * r01 * t1 + r02 * r02 * t2;
  c[1] = r00 * r10 * t0 + r01 * r11 * t1 + r02 * r12 * t2;
  c[2] = r00 * r20 * t0 + r01 * r21 * t1 + r02 * r22 * t2;
  c[3] = c[1];
  c[4] = r10 * r10 * t0 + r11 * r11 * t1 + r12 * r12 * t2;
  c[5] = r10 * r20 * t0 + r11 * r21 * t1 + r12 * r22 * t2;
  c[6] = c[2];
  c[7] = c[5];
  c[8] = r20 * r20 * t0 + r21 * r21 * t1 + r22 * r22 * t2;
}

__global__ __launch_bounds__(TPB) void gauss_cov3d_kernel(
    const float* __restrict__ scales,   // [n,3]
    const v4f*  __restrict__ rots,      // [n,4]
    float* __restrict__ out,            // [n,3,3]
    int n) {
  __shared__ alignas(16) float ldsS[TPB * 3];   // staged scales  (3 KB)
  __shared__ alignas(16) float ldsO[TPB * 9];   // staged cov out (9 KB)

  const int tid  = threadIdx.x;
  const int base = blockIdx.x * TPB;
  const int i    = base + tid;

  if (base + TPB <= n) {
    // ---------- full tile ----------------------------------------------------
    // scales[base*3 .. base*3+768) == 192 x v4f: async DMA straight into LDS,
    // no VGPR staging, tracked on ASYNCcnt.
    const v4f* s4 = reinterpret_cast<const v4f*>(scales + (size_t)base * 3);
    if (tid < (TPB * 3) / 4) {
      async_g2l_b128(s4 + tid, reinterpret_cast<v4f*>(ldsS) + tid);
    }
    // Rotations are consumed by the loading thread itself: direct NT B128.
    v4f q = __builtin_nontemporal_load(rots + i);

    wait_async0();        // my wave's async LDS writes landed
    __syncthreads();      // everyone's landed

    float s0 = ldsS[tid * 3 + 0];
    float s1 = ldsS[tid * 3 + 1];
    float s2 = ldsS[tid * 3 + 2];   // stride-3: bank-conflict free

    float c[9];
    cov_from(q, s0, s1, s2, c);

#pragma unroll
    for (int k = 0; k < 9; ++k) ldsO[tid * 9 + k] = c[k];  // stride-9: conflict free

    __syncthreads();      // all cov rows staged in LDS

    // cov[base*9 .. base*9+2304) == 576 x v4f: async DMA LDS -> global.
    // S_ENDPGM performs an implicit wait-idle, draining ASYNCcnt.
    v4f* o4 = reinterpret_cast<v4f*>(out + (size_t)base * 9);
#pragma unroll
    for (int v = 0; v < (TPB * 9) / 4; v += TPB) {
      int idx = v + tid;
      if (idx < (TPB * 9) / 4) {
        async_l2g_b128(o4 + idx, reinterpret_cast<const v4f*>(ldsO) + idx);
      }
    }
  } else if (i < n) {
    // ---------- scalar tail (block-uniform branch; no barriers here) --------
    float s0 = scales[3 * (size_t)i + 0];
    float s1 = scales[3 * (size_t)i + 1];
    float s2 = scales[3 * (size_t)i + 2];
    v4f q = rots[i];
    float c[9];
    cov_from(q, s0, s1, s2, c);
#pragma unroll
    for (int k = 0; k < 9; ++k) out[(size_t)9 * i + k] = c[k];
  }
}

extern "C" void kernel_launch(void* const* d_in, const int* in_sizes, int n_in,
                              void* d_out, int out_size, void* d_ws, size_t ws_size,
                              hipStream_t stream) {
  const float* scales = (const float*)d_in[0];   // [n,3] f32
  const v4f*   rots   = (const v4f*)d_in[1];     // [n,4] f32
  float*       out    = (float*)d_out;           // [n,3,3] f32

  const int n = in_sizes[0] / 3;
  const int blocks = (n + TPB - 1) / TPB;
  gauss_cov3d_kernel<<<blocks, TPB, 0, stream>>>(scales, rots, out, n);
}